// RouteNTMMatrix_57990648430613
// MI455X (gfx1250) — compile-verified
//
#include <hip/hip_runtime.h>
#include <hip/hip_bf16.h>

#define DIM      256
#define KDIM     100
#define KP       112      // K padded to 7 tiles of 16
#define NT       7
#define CC       5
#define RSTRIDE  264      // 256 + 8 pad -> spreads LDS banks for A-frag reads
#define WAVES    4
#define ROWS_PER_WAVE 16
#define BLOCK_ROWS (WAVES * ROWS_PER_WAVE)   // 64
#define THREADS  (WAVES * 32)                // 128

// ---- LDS layout (float offsets) ----
// w/a/b are stored K-pair interleaved: elem (k, c) -> base + (k>>1)*(2*W) + c*2 + (k&1)
// so a WMMA B-fragment {(kA,c), (kA+1,c)} is one aligned 8B ds_load_b64.
#define OFF_W    0                      // [128][112][2] interleaved w      (28672)
#define OFF_A    28672                  // [128][16][2]  interleaved mode_a (4096)
#define OFF_B    32768                  // [128][16][2]  interleaved mode_b (4096)
#define OFF_FCWG 36864                  // [112] fc_w[0:100] padded with 0
#define OFF_FCWI 36976                  // [32]  fc_w[100:125] padded
#define OFF_BIAS 37008                  // [32]  mode_bias (25) padded
#define OFF_WAVE 37056                  // per-wave regions start
#define PW       8720                   // per-wave size
#define OFF_R    0                      // [16][RSTRIDE]
#define OFF_T    4224                   // [16][RSTRIDE]
#define OFF_RA   8448                   // [16][8]
#define OFF_TB   8576                   // [16][8]
#define OFF_RS   8704                   // [16] per-row gcp.fc_w partial sums
#define LDS_FLOATS (OFF_WAVE + WAVES * PW)   // 71936 floats = 287744 B

typedef __attribute__((ext_vector_type(2))) float v2f;
typedef __attribute__((ext_vector_type(8))) float v8f;

extern __shared__ float lds[];

// 281 KB LDS => exactly one 128-thread workgroup per WGP => 1 wave/SIMD.
// Tell the compiler occupancy-1 is intended so it spends VGPRs on ILP.
__global__ void __launch_bounds__(THREADS, 1)
route_ntm_fused_kernel(const int* __restrict__ route_idx,
                       const int* __restrict__ time_idx,
                       const float* __restrict__ route_table,
                       const float* __restrict__ time_table,
                       const float* __restrict__ w,
                       const float* __restrict__ mode_a,
                       const float* __restrict__ mode_b,
                       const float* __restrict__ mode_bias,
                       const float* __restrict__ fc_w,
                       const float* __restrict__ fc_b,
                       float* __restrict__ out)
{
    const int tid  = threadIdx.x;
    const int lane = tid & 31;
    const int wave = tid >> 5;
    const int half = lane >> 4;     // which 16-lane group (K-split for WMMA frags)
    const int l16  = lane & 15;

    // ---- async gather of this wave's 16 rows of r and t straight into LDS ----
    // GLOBAL_LOAD_ASYNC_TO_LDS_B128: per-lane 16B memory -> LDS, tracked by ASYNCcnt.
    const int rowbase = blockIdx.x * BLOCK_ROWS + wave * ROWS_PER_WAVE;
    {
        const unsigned rbase = (unsigned)(OFF_WAVE + wave * PW + OFF_R) * 4u;
        const unsigned tbase = (unsigned)(OFF_WAVE + wave * PW + OFF_T) * 4u;
        for (int it = 0; it < 2 * ROWS_PER_WAVE; ++it) {
            int m = it >> 1;
            int q = ((it & 1) << 5) + lane;           // float4 index within the row
            int ri = route_idx[rowbase + m];
            int ti = time_idx[rowbase + m];
            const float* rsrc = route_table + (size_t)ri * DIM + q * 4;
            const float* tsrc = time_table  + (size_t)ti * DIM + q * 4;
            unsigned rdst = rbase + (unsigned)(m * RSTRIDE + q * 4) * 4u;
            unsigned tdst = tbase + (unsigned)(m * RSTRIDE + q * 4) * 4u;
            asm volatile("global_load_async_to_lds_b128 %0, %1, off"
                         :: "v"(rdst), "v"(rsrc) : "memory");
            asm volatile("global_load_async_to_lds_b128 %0, %1, off"
                         :: "v"(tdst), "v"(tsrc) : "memory");
        }
    }

    // ---- stage weights into LDS, K-pair interleaved (block-cooperative) ----
    for (int idx = tid; idx < DIM * KP; idx += THREADS) {
        int k = idx / KP, c = idx - k * KP;
        float v = (c < KDIM) ? w[k * KDIM + c] : 0.0f;
        lds[OFF_W + (k >> 1) * (2 * KP) + c * 2 + (k & 1)] = v;
    }
    for (int idx = tid; idx < DIM * 16; idx += THREADS) {
        int k = idx >> 4, c = idx & 15;
        int dst = (k >> 1) * 32 + c * 2 + (k & 1);
        lds[OFF_A + dst] = (c < CC) ? mode_a[k * CC + c] : 0.0f;
        lds[OFF_B + dst] = (c < CC) ? mode_b[k * CC + c] : 0.0f;
    }
    if (tid < KP) lds[OFF_FCWG + tid] = (tid < KDIM) ? fc_w[tid] : 0.0f;
    if (tid < 32) {
        lds[OFF_FCWI + tid] = (tid < CC * CC) ? fc_w[KDIM + tid] : 0.0f;
        lds[OFF_BIAS + tid] = (tid < CC * CC) ? mode_bias[tid] : 0.0f;
    }

    asm volatile("s_wait_asynccnt 0" ::: "memory");   // r/t rows resident in LDS
    __syncthreads();                                   // w/a/b staged by all waves

    // ---- WMMA fp32 GEMMs: (r*t)@w (7 tiles), r@mode_a, t@mode_b ----
    v8f acc[NT];
    v8f accA = {0.f,0.f,0.f,0.f,0.f,0.f,0.f,0.f};
    v8f accB = {0.f,0.f,0.f,0.f,0.f,0.f,0.f,0.f};
#pragma unroll
    for (int nt = 0; nt < NT; ++nt) acc[nt] = accA;

    const float* rl   = &lds[OFF_WAVE + wave * PW + OFF_R];
    const float* tl   = &lds[OFF_WAVE + wave * PW + OFF_T];
    const float* rrow = &rl[l16 * RSTRIDE];
    const float* trow = &tl[l16 * RSTRIDE];

#pragma unroll 4
    for (int kk = 0; kk < DIM / 4; ++kk) {
        const int kA    = kk * 4 + half * 2;  // lanes 0-15: K={4k,4k+1}; 16-31: K={4k+2,4k+3}
        const int kpair = 2 * kk + half;      // interleaved B row-pair index

        // Batch ALL fragment loads for this K-step into distinct registers
        // first: DS returns are in-order, so the scheduler can wait down the
        // DScnt ladder and overlap ds_loads with the matrix pipe.
        v2f ar = *(const v2f*)&rrow[kA];
        v2f at = *(const v2f*)&trow[kA];
        v2f bw[NT];
#pragma unroll
        for (int nt = 0; nt < NT; ++nt)
            bw[nt] = *(const v2f*)&lds[OFF_W + kpair * (2 * KP) + (nt * 16 + l16) * 2];
        v2f ba = *(const v2f*)&lds[OFF_A + kpair * 32 + l16 * 2];
        v2f bb = *(const v2f*)&lds[OFF_B + kpair * 32 + l16 * 2];

        v2f ax = ar * at;
        accA = __builtin_amdgcn_wmma_f32_16x16x4_f32(
            false, ar, false, ba, (short)0, accA, false, false);
        accB = __builtin_amdgcn_wmma_f32_16x16x4_f32(
            false, at, false, bb, (short)0, accB, false, false);
#pragma unroll
        for (int nt = 0; nt < NT; ++nt) {
            acc[nt] = __builtin_amdgcn_wmma_f32_16x16x4_f32(
                false, ax, false, bw[nt], (short)0, acc[nt], false, false);
        }
    }

    // ---- epilogue: C/D layout -> VGPR v holds row M=v (lanes 0-15) / M=v+8 (lanes 16-31), N=lane%16
    float* ral = &lds[OFF_WAVE + wave * PW + OFF_RA];
    float* tbl = &lds[OFF_WAVE + wave * PW + OFF_TB];
    float* rsl = &lds[OFF_WAVE + wave * PW + OFF_RS];

#pragma unroll
    for (int v = 0; v < 8; ++v) {
        int m = v + 8 * half;
        if (l16 < 8) {
            ral[m * 8 + l16] = accA[v];
            tbl[m * 8 + l16] = accB[v];
        }
    }

    // relu(gcp) . fc_w[0:100]  -- padded cols: acc==0 and fcw==0, contribute nothing
    float rowacc[8];
#pragma unroll
    for (int v = 0; v < 8; ++v) rowacc[v] = 0.0f;
#pragma unroll
    for (int nt = 0; nt < NT; ++nt) {
        float fw = lds[OFF_FCWG + nt * 16 + l16];
#pragma unroll
        for (int v = 0; v < 8; ++v) {
            float g = fmaxf(acc[nt][v], 0.0f) * fw;
            g += __shfl_xor(g, 8, 16);
            g += __shfl_xor(g, 4, 16);
            g += __shfl_xor(g, 2, 16);
            g += __shfl_xor(g, 1, 16);
            rowacc[v] += g;                 // full sum present in all 16 lanes of the group
        }
    }
    if (l16 == 0) {
#pragma unroll
        for (int v = 0; v < 8; ++v) rsl[v + 8 * half] = rowacc[v];
    }

    // per-row finish: sigmoid(ra_i*tb_j + bias) . fc_w[100:125] + fc_b
    if (half == 0) {
        int m = l16;
        float ra[CC], tb[CC];
#pragma unroll
        for (int c = 0; c < CC; ++c) { ra[c] = ral[m * 8 + c]; tb[c] = tbl[m * 8 + c]; }
        float s = rsl[m] + fc_b[0];
#pragma unroll
        for (int i = 0; i < CC; ++i) {
#pragma unroll
            for (int j = 0; j < CC; ++j) {
                float z  = ra[i] * tb[j] + lds[OFF_BIAS + i * CC + j];
                float sg = 1.0f / (1.0f + __expf(-z));
                s += sg * lds[OFF_FCWI + i * CC + j];
            }
        }
        out[rowbase + m] = s;
    }
}

extern "C" void kernel_launch(void* const* d_in, const int* in_sizes, int n_in,
                              void* d_out, int out_size, void* d_ws, size_t ws_size,
                              hipStream_t stream) {
    (void)n_in; (void)out_size; (void)d_ws; (void)ws_size;
    const int*   route_idx   = (const int*)  d_in[0];
    const int*   time_idx    = (const int*)  d_in[1];
    const float* route_table = (const float*)d_in[2];
    const float* time_table  = (const float*)d_in[3];
    const float* w           = (const float*)d_in[4];
    const float* mode_a      = (const float*)d_in[5];
    const float* mode_b      = (const float*)d_in[6];
    const float* mode_bias   = (const float*)d_in[7];
    const float* fc_w        = (const float*)d_in[8];
    const float* fc_b        = (const float*)d_in[9];
    float*       out         = (float*)d_out;

    const int batch  = in_sizes[0];
    const int blocks = batch / BLOCK_ROWS;           // 262144/64 = 4096
    const size_t shmem = (size_t)LDS_FLOATS * sizeof(float);  // ~281 KB of the 320 KB WGP LDS

    route_ntm_fused_kernel<<<blocks, THREADS, shmem, stream>>>(
        route_idx, time_idx, route_table, time_table,
        w, mode_a, mode_b, mode_bias, fc_w, fc_b, out);
}